// GNN_20624432955880
// MI455X (gfx1250) — compile-verified
//
#include <hip/hip_runtime.h>

typedef float v2f __attribute__((ext_vector_type(2)));
typedef float v8f __attribute__((ext_vector_type(8)));

static constexpr int HF  = 64;   // hidden width
static constexpr int FIN = 9;    // input features
static constexpr int WP  = 160;  // LDS floats per K-pair row (padded: 160%64==32 -> conflict-free halves)

// ---------------- small init kernels ----------------
__global__ void k_init_deg(float* __restrict__ deg, int N) {
    int i = blockIdx.x * blockDim.x + threadIdx.x;
    if (i < N) deg[i] = 1.0f;               // self-loop contributes 1
}

__global__ void k_zero(float* __restrict__ p, int n) {
    int i = blockIdx.x * blockDim.x + threadIdx.x;
    if (i < n) p[i] = 0.0f;
}

__global__ void k_deg_edges(const int* __restrict__ dst, int E, float* __restrict__ deg) {
    int e = blockIdx.x * blockDim.x + threadIdx.x;
    if (e < E) atomicAdd(&deg[dst[e]], 1.0f);
}

__global__ void k_dis(float* __restrict__ deg, int N) {
    int i = blockIdx.x * blockDim.x + threadIdx.x;
    if (i < N) deg[i] = rsqrtf(fmaxf(deg[i], 1.0f));   // in-place: deg -> deg^{-1/2}
}

// ---------------- layer 1 transform: t = x @ W1 (K=9) ; optional fused self-init ---------
__global__ void k_lin1(const float* __restrict__ x, const float* __restrict__ W1,
                       float* __restrict__ t, float* __restrict__ acc,
                       const float* __restrict__ dis, const float* __restrict__ bias, int N) {
    int idx = blockIdx.x * blockDim.x + threadIdx.x;
    if (idx >= N * HF) return;
    int i = idx >> 6, j = idx & 63;
    float s = 0.0f;
#pragma unroll
    for (int k = 0; k < FIN; ++k) s = fmaf(x[i * FIN + k], W1[k * HF + j], s);
    t[idx] = s;
    if (acc) {                               // acc = t * dis^2 + b  (self-loop + bias)
        float w = dis[i];
        acc[idx] = fmaf(s, w * w, bias[j]);
    }
}

// ---------------- WMMA GEMM: T = relu(A) @ W,  A:[N,64] W:[64,64] ----------------
// One wave computes a 16-row x 64-col stripe with V_WMMA_F32_16X16X4_F32.
// A chunk per lane = one contiguous float2 (lanes 0-15: K=kc,kc+1; lanes 16-31: K=kc+2,kc+3).
// W is staged in LDS swizzled by K-pairs so each B operand is a single aligned ds_load_b64:
//   sW[p*WP + c*2 + (k&1)] = W[k][c],  p = k>>1.
// Optional fused epilogue: ACC = D * dis[row]^2 + bias[col]  (self-loop + bias init).
__global__ void __launch_bounds__(256) k_gemm_relu(const float* __restrict__ A,
                                                   const float* __restrict__ W,
                                                   float* __restrict__ T,
                                                   float* __restrict__ ACC,
                                                   const float* __restrict__ dis,
                                                   const float* __restrict__ bias, int N) {
    __shared__ float sW[32 * WP];
    int tid = threadIdx.x;
    for (int i = tid; i < HF * HF; i += 256) {
        int k = i >> 6, c = i & 63;
        sW[(k >> 1) * WP + c * 2 + (k & 1)] = W[i];
    }
    __syncthreads();

    int wave = tid >> 5;
    int lane = tid & 31;
    int m    = lane & 15;       // row (A) / col (B,D) within tile
    int hi   = lane >> 4;       // 0 or 1
    int koff = hi * 2;

    int rowBase = blockIdx.x * 128 + wave * 16;
    int row  = rowBase + m;
    int rowc = row < N ? row : N - 1;              // clamp so EXEC stays all-ones
    const float* __restrict__ arow = A + (long long)rowc * HF;

    v8f acc0 = {}, acc1 = {}, acc2 = {}, acc3 = {};

#pragma unroll
    for (int kc = 0; kc < HF; kc += 4) {
        float2 av = *(const float2*)(arow + kc + koff);
        v2f a;
        a.x = fmaxf(av.x, 0.0f);                   // fused ReLU on input
        a.y = fmaxf(av.y, 0.0f);

        // B operands: pair-row p = (kc+koff)/2, column = cTile*16 + m  -> one b64 each
        const float* wp = sW + ((kc + koff) >> 1) * WP + m * 2;
        v2f b0 = *(const v2f*)(wp + 0);
        v2f b1 = *(const v2f*)(wp + 32);
        v2f b2 = *(const v2f*)(wp + 64);
        v2f b3 = *(const v2f*)(wp + 96);

        acc0 = __builtin_amdgcn_wmma_f32_16x16x4_f32(false, a, false, b0, (short)0, acc0, false, false);
        acc1 = __builtin_amdgcn_wmma_f32_16x16x4_f32(false, a, false, b1, (short)0, acc1, false, false);
        acc2 = __builtin_amdgcn_wmma_f32_16x16x4_f32(false, a, false, b2, (short)0, acc2, false, false);
        acc3 = __builtin_amdgcn_wmma_f32_16x16x4_f32(false, a, false, b3, (short)0, acc3, false, false);
    }

    float bb0 = 0.f, bb1 = 0.f, bb2 = 0.f, bb3 = 0.f;
    if (ACC) {
        bb0 = bias[m];
        bb1 = bias[m + 16];
        bb2 = bias[m + 32];
        bb3 = bias[m + 48];
    }

    // D layout: VGPR r -> row r (lanes 0-15) / row r+8 (lanes 16-31), col = m
#pragma unroll
    for (int r = 0; r < 8; ++r) {
        int orow = rowBase + r + hi * 8;
        if (orow < N) {
            float* __restrict__ o = T + (long long)orow * HF + m;
            o[0]  = acc0[r];
            o[16] = acc1[r];
            o[32] = acc2[r];
            o[48] = acc3[r];
            if (ACC) {                         // fused self-loop + bias init
                float w = dis[orow];
                w *= w;
                float* __restrict__ ao = ACC + (long long)orow * HF + m;
                ao[0]  = fmaf(acc0[r], w, bb0);
                ao[16] = fmaf(acc1[r], w, bb1);
                ao[32] = fmaf(acc2[r], w, bb2);
                ao[48] = fmaf(acc3[r], w, bb3);
            }
        }
    }
}

// ---------------- fallback self-loop + bias init: acc = t * dis^2 + b ----------------
__global__ void k_self(const float* __restrict__ t, const float* __restrict__ dis,
                       const float* __restrict__ b, float* __restrict__ acc, int N) {
    int idx = blockIdx.x * blockDim.x + threadIdx.x;      // one thread per 2 features
    if (idx >= N * (HF / 2)) return;
    int i = __builtin_amdgcn_readfirstlane(idx >> 5);     // wave-uniform node index -> SGPR
    int f = (idx & 31) * 2;
    float w = dis[i];
    w *= w;
    float2 v = *(const float2*)(t + (long long)i * HF + f);
    float2 o;
    o.x = fmaf(v.x, w, b[f]);
    o.y = fmaf(v.y, w, b[f + 1]);
    *(float2*)(acc + (long long)i * HF + f) = o;
}

// ---------------- edge scatter: acc[dst] += t[src] * (dis[s]*dis[d]) ----------------
// 32 lanes per edge (2 feats each); edge index is wave-uniform -> scalarize all
// per-edge loads (src, dst, both dis values) so vector memory does only the
// feature gather + f32 atomics.
__global__ void k_edges(const int* __restrict__ src, const int* __restrict__ dst,
                        const float* __restrict__ dis, const float* __restrict__ t,
                        float* __restrict__ acc, int E) {
    int idx = blockIdx.x * blockDim.x + threadIdx.x;
    if (idx >= E * (HF / 2)) return;
    int e = __builtin_amdgcn_readfirstlane(idx >> 5);
    int f = (idx & 31) * 2;
    int s = src[e], d = dst[e];
    float nw = dis[s] * dis[d];
    float2 v = *(const float2*)(t + (long long)s * HF + f);
    float* o = acc + (long long)d * HF + f;
    atomicAdd(o,     v.x * nw);
    atomicAdd(o + 1, v.y * nw);
}

// ---------------- mean-pool accumulate: sums[g] += relu(acc[i]), counts[g]++ ----------------
__global__ void k_pool(const float* __restrict__ acc, const int* __restrict__ batch,
                       float* __restrict__ sums, float* __restrict__ counts, int N) {
    int idx = blockIdx.x * blockDim.x + threadIdx.x;
    if (idx >= N * (HF / 2)) return;
    int i = __builtin_amdgcn_readfirstlane(idx >> 5);     // wave-uniform node index
    int f = (idx & 31) * 2;
    int g = batch[i];
    float2 v = *(const float2*)(acc + (long long)i * HF + f);
    float* o = sums + (long long)g * HF + f;
    atomicAdd(o,     fmaxf(v.x, 0.0f));
    atomicAdd(o + 1, fmaxf(v.y, 0.0f));
    if (f == 0) atomicAdd(&counts[g], 1.0f);
}

// ---------------- final: out[g] = (sums[g]/max(cnt,1)) . Wout + bout ----------------
__global__ void k_final(const float* __restrict__ sums, const float* __restrict__ counts,
                        const float* __restrict__ Wout, const float* __restrict__ bout,
                        float* __restrict__ out, int G) {
    int g = blockIdx.x * blockDim.x + threadIdx.x;
    if (g >= G) return;
    float inv = 1.0f / fmaxf(counts[g], 1.0f);
    float s = 0.0f;
#pragma unroll 8
    for (int f = 0; f < HF; ++f) s = fmaf(sums[g * HF + f] * inv, Wout[f], s);
    out[g] = s + bout[0];
}

// ---------------- host launch ----------------
extern "C" void kernel_launch(void* const* d_in, const int* in_sizes, int n_in,
                              void* d_out, int out_size, void* d_ws, size_t ws_size,
                              hipStream_t stream) {
    (void)n_in;
    const float* x     = (const float*)d_in[0];
    const int*   eidx  = (const int*)d_in[1];     // [2, E] flattened
    const int*   batch = (const int*)d_in[2];
    const float* W1 = (const float*)d_in[3];  const float* b1 = (const float*)d_in[4];
    const float* W2 = (const float*)d_in[5];  const float* b2 = (const float*)d_in[6];
    const float* W3 = (const float*)d_in[7];  const float* b3 = (const float*)d_in[8];
    const float* W4 = (const float*)d_in[9];  const float* b4 = (const float*)d_in[10];
    const float* Wout = (const float*)d_in[11]; const float* bout = (const float*)d_in[12];
    float* out = (float*)d_out;

    const int N = in_sizes[0] / FIN;
    const int E = in_sizes[1] / 2;
    const int G = out_size;
    const int* src = eidx;
    const int* dst = eidx + E;

    auto align256 = [](size_t v) { return (v + 255) & ~size_t(255); };
    const size_t szDis  = align256((size_t)N * 4);
    const size_t szBig  = align256((size_t)N * HF * 4);
    const size_t szSums = align256((size_t)G * HF * 4);
    const size_t szCnt  = align256((size_t)G * 4);
    const bool fused = ws_size >= szDis + 3 * szBig + szSums + szCnt;
    const int nBig = fused ? 3 : 2;

    char* base = (char*)d_ws;
    size_t off = 0;
    float* dis = (float*)(base + off); off += szDis;
    float* B[3] = {nullptr, nullptr, nullptr};
    for (int i = 0; i < nBig; ++i) { B[i] = (float*)(base + off); off += szBig; }
    float* sums = (float*)(base + off); off += szSums;
    float* cnts = (float*)(base + off);

    const int TB = 256;
    dim3 blk(TB);
    int gN    = (N + TB - 1) / TB;
    int gE    = (E + TB - 1) / TB;
    int gNH   = (N * HF + TB - 1) / TB;
    int gNH2  = (N * (HF / 2) + TB - 1) / TB;
    int gEH2  = (E * (HF / 2) + TB - 1) / TB;
    int gGemm = (N + 127) / 128;
    int gG    = (G + TB - 1) / TB;

    // degrees -> dis ; zero pooled accumulators
    k_init_deg<<<gN, blk, 0, stream>>>(dis, N);
    k_zero<<<(G * HF + TB - 1) / TB, blk, 0, stream>>>(sums, G * HF);
    k_zero<<<gG, blk, 0, stream>>>(cnts, G);
    k_deg_edges<<<gE, blk, 0, stream>>>(dst, E, dis);
    k_dis<<<gN, blk, 0, stream>>>(dis, N);

    const float* Ws[4] = {W1, W2, W3, W4};
    const float* bs[4] = {b1, b2, b3, b4};

    if (fused) {
        // layer 1: t=B0, acc=B1 (self-init fused)
        k_lin1<<<gNH, blk, 0, stream>>>(x, W1, B[0], B[1], dis, b1, N);
        k_edges<<<gEH2, blk, 0, stream>>>(src, dst, dis, B[0], B[1], E);
        float* in = B[1]; float* tb = B[2]; float* ab = B[0];
        for (int l = 1; l < 4; ++l) {
            k_gemm_relu<<<gGemm, blk, 0, stream>>>(in, Ws[l], tb, ab, dis, bs[l], N);
            k_edges<<<gEH2, blk, 0, stream>>>(src, dst, dis, tb, ab, E);
            float* tmp = in; in = ab; ab = tb; tb = tmp;   // rotate
        }
        k_pool<<<gNH2, blk, 0, stream>>>(in, batch, sums, cnts, N);
    } else {
        float* t = B[0]; float* acc = B[1];
        k_lin1<<<gNH, blk, 0, stream>>>(x, W1, t, nullptr, dis, b1, N);
        for (int l = 0; l < 4; ++l) {
            if (l > 0)
                k_gemm_relu<<<gGemm, blk, 0, stream>>>(acc, Ws[l], t, nullptr, dis, bs[l], N);
            k_self<<<gNH2, blk, 0, stream>>>(t, dis, bs[l], acc, N);
            k_edges<<<gEH2, blk, 0, stream>>>(src, dst, dis, t, acc, E);
        }
        k_pool<<<gNH2, blk, 0, stream>>>(acc, batch, sums, cnts, N);
    }

    k_final<<<gG, blk, 0, stream>>>(sums, cnts, Wout, bout, out, G);
}